// MultiHeadEMABlock_44779329028677
// MI455X (gfx1250) — compile-verified
//
#include <hip/hip_runtime.h>
#include <math.h>

// MultiHeadEMA fused kernel for MI455X (gfx1250, wave32).
//
// out[b,d,t] = x[b,d,t] + sum_h w[h,d] * EMA_h(LN(x)[b,:,d])[t],  w = E*R elementwise.
// EMA chunked into 16-step blocks:  Y = K_h @ X + qpow(x)y_in, five chained
// V_WMMA_F32_16X16X4_F32 per head per 16-step chunk (4 K-slices + 1 carry).
//
// Memory-bound (~96MB total traffic -> ~4.1us at 23.3 TB/s). The serial scan is
// latency-hidden via global_prefetch_b8 (4 iters = 256 timesteps ahead) and
// 64-timestep iterations (2 barriers / 4 KB of traffic per iteration per WG).

#define B_   4
#define DIM  512
#define N_   4096
#define H_   8
#define EPS_ 1e-5f
#define TPI  64    // timesteps per iteration = 4 WMMA sub-chunks
#define LDSX 68    // LDS row stride (floats): 272B, 16B-aligned; 4j mod 64 distinct
#define PF   4     // prefetch distance in iterations (256 timesteps ahead)

typedef __attribute__((ext_vector_type(2))) float v2f;
typedef __attribute__((ext_vector_type(8))) float v8f;

__device__ __forceinline__ float sigmf(float v) { return 1.0f / (1.0f + __expf(-v)); }

// ---------------- Kernel 1: LayerNorm statistics (mean, rstd per (b,t)) ----------------
__global__ void __launch_bounds__(256) ln_stats_kernel(const float* __restrict__ x,
                                                       float* __restrict__ mean,
                                                       float* __restrict__ rstd) {
  int b    = blockIdx.x >> 6;          // N_/64 = 64 t-tiles per batch
  int t0   = (blockIdx.x & 63) << 6;
  int tl   = threadIdx.x & 63;
  int part = threadIdx.x >> 6;

  const float* xb = x + (size_t)b * DIM * N_ + t0 + tl;
  float s = 0.0f, s2 = 0.0f;
#pragma unroll 8
  for (int c = part; c < DIM; c += 4) {
    float v = xb[(size_t)c * N_];
    s += v; s2 += v * v;
  }
  __shared__ float sh[2][4][64];
  sh[0][part][tl] = s;
  sh[1][part][tl] = s2;
  __syncthreads();
  if (part == 0) {
    s  = sh[0][0][tl] + sh[0][1][tl] + sh[0][2][tl] + sh[0][3][tl];
    s2 = sh[1][0][tl] + sh[1][1][tl] + sh[1][2][tl] + sh[1][3][tl];
    float m   = s * (1.0f / DIM);
    float var = s2 * (1.0f / DIM) - m * m;
    mean[b * N_ + t0 + tl] = m;
    rstd[b * N_ + t0 + tl] = rsqrtf(var + EPS_);
  }
}

// ---------------- Kernel 2: fused normalize + 8-head EMA (WMMA) + combine + residual ----
// Grid: B_*(DIM/16) = 128 blocks, 256 threads = 8 waves; wave id == head id.
// 64 iterations of 64 timesteps; per-wave carry propagated through 16-step sub-chunks.
__global__ void __launch_bounds__(256)
ema_wmma_kernel(const float* __restrict__ x,
                const float* __restrict__ gamma,
                const float* __restrict__ beta,
                const float* __restrict__ expansion,
                const float* __restrict__ reduction,
                const float* __restrict__ alphas,
                const float* __restrict__ damp,
                const float* __restrict__ mean,
                const float* __restrict__ rstd,
                float* __restrict__ out) {
  const int b    = blockIdx.x >> 5;        // DIM/16 = 32 channel tiles
  const int d0   = (blockIdx.x & 31) << 4;
  const int tid  = threadIdx.x;
  const int h    = tid >> 5;               // wave == head
  const int lane = tid & 31;
  const int j    = lane & 15;              // N (channel) index; M index for A rows
  const int half = lane >> 4;

  const float a = sigmf(alphas[h]);
  const float q = (1.0f - a) * sigmf(damp[h]);
  const float w = expansion[h * DIM + d0 + j] * reduction[h * DIM + d0 + j];

  // A operand layout for V_WMMA_F32_16X16X4_F32 (ISA 7.12.2):
  //   component r, lanes<16: (M=lane, K=r);  lanes>=16: (M=lane-16, K=r+2).
  // K_h[i,s] = a * q^(i-s) for i>=s else 0, as 4 K-slices of width 4.
  v2f A[4];
#pragma unroll
  for (int k = 0; k < 4; ++k)
#pragma unroll
    for (int r = 0; r < 2; ++r) {
      int col = 4 * k + r + 2 * half;
      A[k][r] = (j >= col) ? a * __powf(q, (float)(j - col)) : 0.0f;
    }
  // Carry matrix: column 0 = q^(M+1), other columns zero.
  v2f Ac;
  Ac[0] = (half == 0) ? __powf(q, (float)(j + 1)) : 0.0f;
  Ac[1] = 0.0f;

  __shared__ __align__(16) float lds_x[16 * LDSX];    // normalized tile [d_row][t 0..63]
  __shared__ __align__(16) float lds_acc[16 * LDSX];  // head-combined output tile

  for (int i = tid; i < 16 * LDSX; i += 256) lds_acc[i] = 0.0f;

  // Stage/drain mapping: thread -> (row = tid>>4, 4 floats at col (tid&15)*4).
  const int srow = tid >> 4;
  const int scol = (tid & 15) << 2;
  const size_t rowbase = (size_t)(b * DIM + d0) * N_;
  const float* xrow = x + rowbase + (size_t)srow * N_;
  float*       orow = out + rowbase + (size_t)srow * N_;
  const float* mrow = mean + b * N_;
  const float* rrow = rstd + b * N_;
  const float g  = gamma[d0 + srow];
  const float bt = beta[d0 + srow];

  float ycarry = 0.0f;

  for (int it = 0; it < N_ / TPI; ++it) {
    const int t0 = it * TPI;

    // ---- Stage: coalesced raw load (kept for residual), normalize, store to LDS ----
    const float4 xr = *(const float4*)(xrow + t0 + scol);
    const float4 m0 = *(const float4*)(mrow + t0 + scol);
    const float4 r0 = *(const float4*)(rrow + t0 + scol);
    float4 n;
    n.x = (xr.x - m0.x) * r0.x * g + bt;
    n.y = (xr.y - m0.y) * r0.y * g + bt;
    n.z = (xr.z - m0.z) * r0.z * g + bt;
    n.w = (xr.w - m0.w) * r0.w * g + bt;
    *(float4*)&lds_x[srow * LDSX + scol] = n;

    // ---- Prefetch future iterations into WGP$/L2 (hides HBM latency of the scan) ----
    if (it + PF < N_ / TPI)
      __builtin_prefetch(xrow + t0 + PF * TPI + scol, 0, 3);

    __syncthreads();  // lds_x staged; lds_acc zeroed (prev drain / init)

    // ---- Per-head EMA over four 16-step sub-chunks (serial carry) ----
#pragma unroll
    for (int sub = 0; sub < 4; ++sub) {
      const int s0 = sub * 16;
      // B operands (4x16 f32, dual of A): component r, lane -> (K=r+2*half, N=j);
      // X[s, n] = lds_x[n*LDSX + s0 + s]. Bank-conflict-free (4j mod 64 distinct).
      v2f Bk[4];
#pragma unroll
      for (int k = 0; k < 4; ++k)
#pragma unroll
        for (int r = 0; r < 2; ++r)
          Bk[k][r] = lds_x[j * LDSX + s0 + 4 * k + r + 2 * half];
      v2f By;
      By[0] = (half == 0) ? ycarry : 0.0f;  // row 0 of B = y_in[N]
      By[1] = 0.0f;

      v8f acc = {0.f, 0.f, 0.f, 0.f, 0.f, 0.f, 0.f, 0.f};
      acc = __builtin_amdgcn_wmma_f32_16x16x4_f32(false, Ac, false, By, (short)0, acc, false, false);
#pragma unroll
      for (int k = 0; k < 4; ++k)
        acc = __builtin_amdgcn_wmma_f32_16x16x4_f32(false, A[k], false, Bk[k], (short)0, acc, false, false);

      // New carry = Y row M=15 (VGPR 7, lanes 16..31 hold N=lane-16).
      ycarry = __shfl(acc[7], 16 + j, 32);

      // Combine heads: lds_acc[d=N][t=s0+M] += w * Y  (ds_add_f32).
#pragma unroll
      for (int v = 0; v < 8; ++v)
        atomicAdd(&lds_acc[j * LDSX + s0 + v + 8 * half], w * acc[v]);
    }

    __syncthreads();  // all heads' adds complete

    // ---- Drain: combined tile + residual (raw x still in registers), re-zero ----
    float4 o = *(float4*)&lds_acc[srow * LDSX + scol];
    const float4 z = {0.f, 0.f, 0.f, 0.f};
    *(float4*)&lds_acc[srow * LDSX + scol] = z;
    o.x += xr.x; o.y += xr.y; o.z += xr.z; o.w += xr.w;
    *(float4*)(orow + t0 + scol) = o;
  }
}

extern "C" void kernel_launch(void* const* d_in, const int* in_sizes, int n_in,
                              void* d_out, int out_size, void* d_ws, size_t ws_size,
                              hipStream_t stream) {
  (void)in_sizes; (void)n_in; (void)out_size; (void)ws_size;
  const float* x         = (const float*)d_in[0];
  const float* ln_gamma  = (const float*)d_in[1];
  const float* ln_beta   = (const float*)d_in[2];
  const float* expansion = (const float*)d_in[3];
  const float* reduction = (const float*)d_in[4];
  const float* alphas    = (const float*)d_in[5];
  const float* dampen    = (const float*)d_in[6];
  float* out  = (float*)d_out;
  float* mean = (float*)d_ws;            // [B_, N_]
  float* rstd = mean + B_ * N_;          // [B_, N_]   (128 KB total workspace)

  ln_stats_kernel<<<B_ * (N_ / 64), 256, 0, stream>>>(x, mean, rstd);
  ema_wmma_kernel<<<B_ * (DIM / 16), 256, 0, stream>>>(
      x, ln_gamma, ln_beta, expansion, reduction, alphas, dampen, mean, rstd, out);
}